// KANCubic1D_4037269258293
// MI455X (gfx1250) — compile-verified
//
#include <hip/hip_runtime.h>
#include <stdint.h>

// Problem constants (from the reference)
#define BB      32
#define CC      192
#define HH      64
#define WW      64
#define KK      32
#define PLANE   (HH * WW)      // 4096 elements = 16 KB per (b,c) plane
#define THREADS 256            // 8 wave32s

typedef unsigned int v4u __attribute__((ext_vector_type(4)));
typedef int          v4i __attribute__((ext_vector_type(4)));
typedef int          v8i __attribute__((ext_vector_type(8)));

// ---------------------------------------------------------------------------
// TDM: DMA one contiguous 4096-float tile from global memory into LDS.
// D# layout per CDNA5 ISA ch.8 (groups 0 and 1; groups 2/3 zeroed = unused).
// ---------------------------------------------------------------------------
__device__ __forceinline__ void tdm_load_plane_to_lds(const float* gsrc,
                                                      uint32_t lds_byte_addr)
{
    const uint64_t gaddr = (uint64_t)(uintptr_t)gsrc;

    v4u g0;
    g0.x = 1u;                                        // count=1 (valid user D#)
    g0.y = lds_byte_addr;                             // lds_addr (bytes)
    g0.z = (uint32_t)(gaddr & 0xFFFFFFFFu);           // global_addr[31:0]
    g0.w = (uint32_t)((gaddr >> 32) & 0x01FFFFFFu)    // global_addr[56:32]
         | 0x80000000u;                               // type=2 ("image")

    v8i g1;
    g1[0] = (int)(2u << 16);                          // wgmask=0, data_size=2 (4 B)
    g1[1] = (int)(((uint32_t)PLANE & 0xFFFFu) << 16); // tensor_dim0[15:0] -> bits[63:48]
    g1[2] = (int)(((uint32_t)PLANE >> 16) | (1u << 16)); // tensor_dim0[31:16] | tensor_dim1=1
    g1[3] = (int)((uint32_t)PLANE << 16);             // tensor_dim1[31:16]=0 | tile_dim0=4096
    g1[4] = 1;                                        // tile_dim1=1, tile_dim2=0
    g1[5] = PLANE;                                    // tensor_dim0_stride[31:0]
    g1[6] = 0;                                        // stride0[47:32] | stride1[15:0]
    g1[7] = 0;                                        // stride1[47:16]

    v4i g2 = {0, 0, 0, 0};
    v4i g3 = {0, 0, 0, 0};

#if __has_builtin(__builtin_amdgcn_tensor_load_to_lds)
  #if __clang_major__ >= 23
    v8i g4 = {0, 0, 0, 0, 0, 0, 0, 0};
    __builtin_amdgcn_tensor_load_to_lds(g0, g1, g2, g3, g4, 0);
  #else
    __builtin_amdgcn_tensor_load_to_lds(g0, g1, g2, g3, 0);
  #endif
#else
    asm volatile("tensor_load_to_lds %0, %1, %2, %3"
                 :: "s"(g0), "s"(g1), "s"(g2), "s"(g3) : "memory");
#endif
}

__device__ __forceinline__ void wait_tensorcnt0()
{
#if __has_builtin(__builtin_amdgcn_s_wait_tensorcnt)
    __builtin_amdgcn_s_wait_tensorcnt(0);
#else
    asm volatile("s_wait_tensorcnt 0x0" ::: "memory");
#endif
}

// ---------------------------------------------------------------------------
// One workgroup per (b,c) plane:
//   1) wave0 issues the TDM load of the plane into LDS immediately,
//   2) all waves fill the padded alpha table / load scalars (DMA in flight),
//   3) wave0 waits TENSORcnt, workgroup barrier,
//   4) compute from LDS (ds_load_b128 + 4x ds_load_b32 gather), float4 stores.
// ---------------------------------------------------------------------------
extern "C" __global__ void __launch_bounds__(THREADS)
kan_cubic_kernel(const float* __restrict__ x,
                 const float* __restrict__ a,
                 const float* __restrict__ b,
                 const float* __restrict__ alpha,
                 const float* __restrict__ gain,
                 const float* __restrict__ bias,
                 float* __restrict__ out)
{
    __shared__ float tile[PLANE];   // 16 KB staged input plane
    __shared__ float atab[52];      // padded alpha row: atab[j] = alpha[c][clamp(j-9,0,31)]

    const int    p    = blockIdx.x;        // plane index in [0, B*C)
    const int    c    = p % CC;
    const size_t base = (size_t)p * PLANE;

    // Kick off the DMA as early as possible (wave 0 only; waves 1..7 skip via execz).
    if (threadIdx.x < 32) {
        tdm_load_plane_to_lds(x + base, (uint32_t)(uintptr_t)(&tile[0]));
    }

    // Overlap: per-channel scalars + padded alpha table while the DMA runs.
    const float sa    = a[c];
    const float sb    = b[c];
    const float sg    = gain[c];
    const float sbias = bias[c];

    // i in [-8,38] => atab index i+8 in [0,46]; +3 taps => max 49.
    for (int j = threadIdx.x; j < 50; j += THREADS) {
        int k = j - 9;
        k = k < 0 ? 0 : (k > (KK - 1) ? (KK - 1) : k);
        atab[j] = alpha[c * KK + k];
    }

    if (threadIdx.x < 32) {
        wait_tensorcnt0();          // issuing wave drains its TENSORcnt
    }
    __syncthreads();                // LDS tile + atab now visible to all waves

    const float4* __restrict__ src = (const float4*)tile;
    float4*       __restrict__ dst = (float4*)(out + base);

    #pragma unroll
    for (int r = 0; r < PLANE / (4 * THREADS); ++r) {     // 4 iterations
        const int idx = r * THREADS + (int)threadIdx.x;
        float4 v = src[idx];                              // ds_load_b128
        float in[4]  = {v.x, v.y, v.z, v.w};
        float res[4];

        #pragma unroll
        for (int e = 0; e < 4; ++e) {
            const float xv = in[e];
            float xa = fmaf(xv, sa, sb);
            xa = fminf(fmaxf(xa, -1.5f), 1.5f);           // v_med3_f32
            const float u  = fmaf(xa, 15.5f, 15.5f);      // [-1,1] -> [0,31]
            const float fi = floorf(u);
            const float t  = u - fi;                      // exactly in [0,1) in f32
            const int   ii = (int)fi + 8;                 // padded-table base (no clamps)

            const float omt = 1.0f - t;
            const float t2  = t * t;
            const float t3  = t2 * t;
            const float w0  = omt * omt * omt * (1.0f / 6.0f);
            const float w1  = fmaf(3.0f, t3, fmaf(-6.0f, t2, 4.0f)) * (1.0f / 6.0f);
            const float w2  = fmaf(3.0f, (t + t2) - t3, 1.0f) * (1.0f / 6.0f);
            const float w3  = t3 * (1.0f / 6.0f);

            const float a0 = atab[ii + 0];
            const float a1 = atab[ii + 1];
            const float a2 = atab[ii + 2];
            const float a3 = atab[ii + 3];

            // bias folded into the innermost accumulator of the spline chain
            const float spline = fmaf(a0, w0, fmaf(a1, w1, fmaf(a2, w2, fmaf(a3, w3, sbias))));
            res[e] = fmaf(sg, xv, spline);
        }

        dst[idx] = make_float4(res[0], res[1], res[2], res[3]);  // global_store_b128
    }
}

// ---------------------------------------------------------------------------
extern "C" void kernel_launch(void* const* d_in, const int* in_sizes, int n_in,
                              void* d_out, int out_size, void* d_ws, size_t ws_size,
                              hipStream_t stream)
{
    const float* x     = (const float*)d_in[0];
    const float* a     = (const float*)d_in[1];
    const float* b     = (const float*)d_in[2];
    const float* alpha = (const float*)d_in[3];
    const float* gain  = (const float*)d_in[4];
    const float* bias  = (const float*)d_in[5];
    float*       out   = (float*)d_out;

    (void)in_sizes; (void)n_in; (void)out_size; (void)d_ws; (void)ws_size;

    dim3 grid(BB * CC);   // 6144 planes
    dim3 block(THREADS);
    kan_cubic_kernel<<<grid, block, 0, stream>>>(x, a, b, alpha, gain, bias, out);
}